// Gemma3nAltUp_22797686407820
// MI455X (gfx1250) — compile-verified
//
#include <hip/hip_runtime.h>
#include <math.h>

typedef __attribute__((ext_vector_type(2))) float v2f;
typedef __attribute__((ext_vector_type(8))) float v8f;

#define HDIM 2048
#define H4   512      // float4 chunks per row
#define NTOK 16       // tokens per workgroup (one per wave32)
#define NTHR 512      // 16 waves
#define EPSV 1e-6f

__device__ __forceinline__ float wave_reduce_add(float v) {
#pragma unroll
    for (int off = 16; off >= 1; off >>= 1)
        v += __shfl_xor(v, off, 32);
    return v;
}

__device__ __forceinline__ float4 fma4(float4 a, float s, float4 c) {
    float4 r;
    r.x = fmaf(a.x, s, c.x);
    r.y = fmaf(a.y, s, c.y);
    r.z = fmaf(a.z, s, c.z);
    r.w = fmaf(a.w, s, c.w);
    return r;
}

__global__ __launch_bounds__(NTHR)
void altup_fused(const float* __restrict__ hs,      // [4, T, H]
                 const float* __restrict__ act,     // [T, H]
                 const float* __restrict__ wnorm,   // [H]
                 const float* __restrict__ Wrouter, // [4, H]
                 const float* __restrict__ Wpred,   // [16, 4]
                 const float* __restrict__ Wcorr,   // [4, 4]
                 float* __restrict__ out,           // [4, T, H]
                 int T)
{
    __shared__ __align__(16) float s_wr[4 * HDIM];   // w_norm * W_router, 32 KB
    __shared__ float s_modsA[NTOK][4];
    __shared__ float s_modsC[NTOK][4];
    __shared__ float s_coef[NTOK][16];
    __shared__ float s_cc[NTOK][4];

    const int tid  = threadIdx.x;
    const int lane = tid & 31;
    const int wav  = tid >> 5;
    const int tok  = blockIdx.x * NTOK + wav;
    const bool tok_ok = (tok < T);

    // ---- stage w_norm * W_router into LDS (layout identical to W_router) ----
#pragma unroll
    for (int k = 0; k < (4 * HDIM) / NTHR; ++k) {
        int idx = tid + k * NTHR;
        s_wr[idx] = wnorm[idx & (HDIM - 1)] * Wrouter[idx];
    }
    __syncthreads();

    // ---- phase A: per-token reductions (sumsq + 4 router dots, x2 inputs) ----
    float s0 = 0.f, s1 = 0.f;
    float d0[4] = {0.f, 0.f, 0.f, 0.f};
    float d1[4] = {0.f, 0.f, 0.f, 0.f};
    if (tok_ok) {
        const float4* x4 = (const float4*)hs  + (size_t)tok * H4;   // plane 0
        const float4* a4 = (const float4*)act + (size_t)tok * H4;
        const float4* w4 = (const float4*)s_wr;
#pragma unroll 4
        for (int it = 0; it < H4 / 32; ++it) {
            int h4 = lane + it * 32;
            float4 x = x4[h4];
            float4 a = a4[h4];
            s0 = fmaf(x.x, x.x, fmaf(x.y, x.y, fmaf(x.z, x.z, fmaf(x.w, x.w, s0))));
            s1 = fmaf(a.x, a.x, fmaf(a.y, a.y, fmaf(a.z, a.z, fmaf(a.w, a.w, s1))));
#pragma unroll
            for (int r = 0; r < 4; ++r) {
                float4 w = w4[r * H4 + h4];
                d0[r] = fmaf(x.x, w.x, fmaf(x.y, w.y, fmaf(x.z, w.z, fmaf(x.w, w.w, d0[r]))));
                d1[r] = fmaf(a.x, w.x, fmaf(a.y, w.y, fmaf(a.z, w.z, fmaf(a.w, w.w, d1[r]))));
            }
        }
    }
    s0 = wave_reduce_add(s0);
    s1 = wave_reduce_add(s1);
#pragma unroll
    for (int r = 0; r < 4; ++r) {
        d0[r] = wave_reduce_add(d0[r]);
        d1[r] = wave_reduce_add(d1[r]);
    }
    if (tok_ok && lane == 0) {
        const float scale = 1.0f / (float)HDIM;            // ROUTER_INPUT_SCALE
        float inv0 = rsqrtf(s0 * scale + EPSV) * scale;    // rsqrt(mean sq + eps) * H^-1
        float inv1 = rsqrtf(s1 * scale + EPSV) * scale;
#pragma unroll
        for (int r = 0; r < 4; ++r) {
            s_modsA[wav][r] = tanhf(d0[r] * inv0);
            s_modsC[wav][r] = tanhf(d1[r] * inv1);
        }
    }
    __syncthreads();

    // ---- phase B: tiny matmuls on the matrix pipe, wave 0 only ----
    // D[16tok x 16coef] = A[16tok x K4](mods) * B[K4 x 16](W_pred^T)
    // A f32 16x4 layout: lanes 0-15 hold K={0,1}, lanes 16-31 hold K={2,3} (M = lane%16)
    // B mirrors with N = lane%16.  D: VGPR v, lane l -> token v+8*(l>=16), coef l%16.
    if (wav == 0) {
        const int m    = lane & 15;
        const int half = lane >> 4;

        v2f a, b;
        a.x = s_modsA[m][2 * half + 0];
        a.y = s_modsA[m][2 * half + 1];
        b.x = Wpred[m * 4 + 2 * half + 0];   // B[k][j] = W_pred[j][k]
        b.y = Wpred[m * 4 + 2 * half + 1];
        v8f c = {};
        v8f d = __builtin_amdgcn_wmma_f32_16x16x4_f32(
            false, a, false, b, (short)0, c, false, false);
#pragma unroll
        for (int v = 0; v < 8; ++v)
            s_coef[v + half * 8][m] = d[v];

        v2f a2, b2;
        a2.x = s_modsC[m][2 * half + 0];
        a2.y = s_modsC[m][2 * half + 1];
        int mm = (m < 4) ? m : 0;
        b2.x = (m < 4) ? Wcorr[mm * 4 + 2 * half + 0] : 0.0f;  // B[k][j] = W_corr[j][k], padded
        b2.y = (m < 4) ? Wcorr[mm * 4 + 2 * half + 1] : 0.0f;
        v8f d2 = __builtin_amdgcn_wmma_f32_16x16x4_f32(
            false, a2, false, b2, (short)0, c, false, false);
        if (m < 4) {
#pragma unroll
            for (int v = 0; v < 8; ++v)
                s_cc[v + half * 8][m] = d2[v] + 1.0f;          // ccoefs = . + 1
        }
    }
    __syncthreads();

    // ---- phase C: fused predict+correct elementwise pass ----
    if (tok_ok) {
        float cf[16], cc[4];
#pragma unroll
        for (int j = 0; j < 16; ++j) cf[j] = s_coef[wav][j];
#pragma unroll
        for (int j = 0; j < 4; ++j)  cc[j] = s_cc[wav][j];

        const size_t plane4 = (size_t)T * H4;
        const float4* p0 = (const float4*)hs + (size_t)tok * H4;
        const float4* p1 = p0 + plane4;
        const float4* p2 = p1 + plane4;
        const float4* p3 = p2 + plane4;
        const float4* pa = (const float4*)act + (size_t)tok * H4;
        float4* o0 = (float4*)out + (size_t)tok * H4;
        float4* o1 = o0 + plane4;
        float4* o2 = o1 + plane4;
        float4* o3 = o2 + plane4;

#pragma unroll 2
        for (int it = 0; it < H4 / 32; ++it) {
            int h4 = lane + it * 32;
            float4 x0 = p0[h4];
            float4 x1 = p1[h4];
            float4 x2 = p2[h4];
            float4 x3 = p3[h4];
            float4 a  = pa[h4];

            // pred_j = x_j + sum_k x_k * raw[4j+k]
            float4 q0 = fma4(x0, cf[0],  fma4(x1, cf[1],  fma4(x2, cf[2],  fma4(x3, cf[3],  x0))));
            float4 q1 = fma4(x0, cf[4],  fma4(x1, cf[5],  fma4(x2, cf[6],  fma4(x3, cf[7],  x1))));
            float4 q2 = fma4(x0, cf[8],  fma4(x1, cf[9],  fma4(x2, cf[10], fma4(x3, cf[11], x2))));
            float4 q3 = fma4(x0, cf[12], fma4(x1, cf[13], fma4(x2, cf[14], fma4(x3, cf[15], x3))));

            float4 inn;
            inn.x = a.x - q0.x;
            inn.y = a.y - q0.y;
            inn.z = a.z - q0.z;
            inn.w = a.w - q0.w;

            o0[h4] = fma4(inn, cc[0], q0);
            o1[h4] = fma4(inn, cc[1], q1);
            o2[h4] = fma4(inn, cc[2], q2);
            o3[h4] = fma4(inn, cc[3], q3);
        }
    }
}

extern "C" void kernel_launch(void* const* d_in, const int* in_sizes, int n_in,
                              void* d_out, int out_size, void* d_ws, size_t ws_size,
                              hipStream_t stream) {
    const float* hs      = (const float*)d_in[0];  // [4,B,S,H]
    const float* act     = (const float*)d_in[1];  // [B,S,H]
    const float* wnorm   = (const float*)d_in[2];  // [H]
    const float* Wrouter = (const float*)d_in[3];  // [4,H]
    const float* Wpred   = (const float*)d_in[4];  // [16,4]
    const float* Wcorr   = (const float*)d_in[5];  // [4,4]
    float* out = (float*)d_out;                    // [4,B,S,H]

    const int T = in_sizes[1] / HDIM;              // B*S tokens
    const int blocks = (T + NTOK - 1) / NTOK;
    altup_fused<<<blocks, NTHR, 0, stream>>>(hs, act, wnorm, Wrouter, Wpred, Wcorr, out, T);
}